// QNetwork_89627377533571
// MI455X (gfx1250) — compile-verified
//
#include <hip/hip_runtime.h>

typedef __attribute__((ext_vector_type(2))) float v2f;
typedef __attribute__((ext_vector_type(8))) float v8f;

#define HID 32

// ---------------------------------------------------------------------------
// D = A(16x4) * B(4x16) + C, fp32 WMMA (CDNA5 V_WMMA_F32_16X16X4_F32)
// ---------------------------------------------------------------------------
__device__ __forceinline__ v8f wmma_f32(v2f a, v2f b, v8f c) {
  // 8 args: (neg_a, A, neg_b, B, c_mod, C, reuse_a, reuse_b)
  return __builtin_amdgcn_wmma_f32_16x16x4_f32(false, a, false, b, (short)0, c,
                                               false, false);
}

// ---------------------------------------------------------------------------
// Degree count via float atomics, then invert (mean = sum / max(deg,1))
// ---------------------------------------------------------------------------
__global__ void degree_kernel(const int* __restrict__ dst,
                              float* __restrict__ deg, int E) {
  int i = blockIdx.x * blockDim.x + threadIdx.x;
  if (i < E) atomicAdd(&deg[dst[i]], 1.0f);
}

__global__ void invdeg_kernel(float* __restrict__ deg, int n) {
  int i = blockIdx.x * blockDim.x + threadIdx.x;
  if (i < n) deg[i] = 1.0f / fmaxf(deg[i], 1.0f);
}

// ---------------------------------------------------------------------------
// Edge scatter: msg[dst] += x[src], one thread per (edge, feature).
// For DIM=32 one wave == one edge: coalesced 128B gather + atomic burst.
// ---------------------------------------------------------------------------
template <int DIM>
__global__ void scatter_kernel(const float* __restrict__ xin,
                               const int* __restrict__ src,
                               const int* __restrict__ dst,
                               float* __restrict__ msg, int E) {
  unsigned idx = blockIdx.x * blockDim.x + threadIdx.x;
  if (idx >= (unsigned)E * DIM) return;
  unsigned e = idx / DIM;
  unsigned k = idx % DIM;
  unsigned s = (unsigned)src[e];
  unsigned d = (unsigned)dst[e];
  atomicAdd(&msg[(size_t)d * DIM + k], xin[(size_t)s * DIM + k]);
}

// ---------------------------------------------------------------------------
// SAGE node update: xout = relu( (msg*invdeg) @ Wl + bl + xin @ Wr )
// One wave per 16-node tile; two 16x16 f32 accumulators cover HID=32 cols.
// Fragment layouts per CDNA5 ISA 7.12.2 (f32 A 16x4, B 4x16, C/D 16x16).
// ---------------------------------------------------------------------------
template <int DIN>
__global__ void sage_gemm_kernel(const float* __restrict__ xin,
                                 const float* __restrict__ msg,
                                 const float* __restrict__ invdeg,
                                 const float* __restrict__ Wl,
                                 const float* __restrict__ bl,
                                 const float* __restrict__ Wr,
                                 float* __restrict__ xout, int n) {
  const int lane = threadIdx.x & 31;
  const int wave = threadIdx.x >> 5;
  const int half = lane >> 4;   // 0: K0/K1 (A) rows 0-7 (D); 1: K2/K3, rows 8-15
  const int lm   = lane & 15;
  const int node0 = (blockIdx.x * (blockDim.x >> 5) + wave) * 16;
  if (node0 >= n) return;  // wave-uniform: EXEC stays all-ones for WMMA

  const int row = node0 + lm;          // A-matrix row held by this lane
  const float rdeg = invdeg[row];
  const float c0 = bl[lm];
  const float c1 = bl[lm + 16];
  v8f acc0 = {c0, c0, c0, c0, c0, c0, c0, c0};
  v8f acc1 = {c1, c1, c1, c1, c1, c1, c1, c1};

  // mean @ Wl
#pragma unroll
  for (int k0 = 0; k0 < DIN; k0 += 4) {
    const int ka = k0 + 2 * half;
    v2f a = *(const v2f*)(msg + (size_t)row * DIN + ka);
    a.x *= rdeg;
    a.y *= rdeg;
    v2f b0 = {Wl[ka * HID + lm],      Wl[(ka + 1) * HID + lm]};
    v2f b1 = {Wl[ka * HID + lm + 16], Wl[(ka + 1) * HID + lm + 16]};
    acc0 = wmma_f32(a, b0, acc0);
    acc1 = wmma_f32(a, b1, acc1);
  }
  // xin @ Wr
#pragma unroll
  for (int k0 = 0; k0 < DIN; k0 += 4) {
    const int ka = k0 + 2 * half;
    v2f a = *(const v2f*)(xin + (size_t)row * DIN + ka);
    v2f b0 = {Wr[ka * HID + lm],      Wr[(ka + 1) * HID + lm]};
    v2f b1 = {Wr[ka * HID + lm + 16], Wr[(ka + 1) * HID + lm + 16]};
    acc0 = wmma_f32(a, b0, acc0);
    acc1 = wmma_f32(a, b1, acc1);
  }

  // D layout: VGPR v, lanes<16 -> row v; lanes>=16 -> row v+8; col = lm(+16)
#pragma unroll
  for (int v = 0; v < 8; v++) {
    const int m = node0 + v + 8 * half;
    xout[(size_t)m * HID + lm]      = fmaxf(acc0[v], 0.0f);
    xout[(size_t)m * HID + lm + 16] = fmaxf(acc1[v], 0.0f);
  }
}

// ---------------------------------------------------------------------------
// Head: out = relu(h @ Wh1 + bh1) @ Wh2 + bh2   (Wh2: 32x3)
// WMMA for the 32x32 GEMM, relu -> per-wave LDS, scalar tail for 3 cols.
// ---------------------------------------------------------------------------
__global__ void head_kernel(const float* __restrict__ h,
                            const float* __restrict__ W1,
                            const float* __restrict__ b1,
                            const float* __restrict__ W2,
                            const float* __restrict__ b2,
                            float* __restrict__ out, int n) {
  __shared__ float tbuf[8][16 * HID];  // per-wave 16x32 tile
  const int lane = threadIdx.x & 31;
  const int wave = threadIdx.x >> 5;
  const int half = lane >> 4;
  const int lm   = lane & 15;
  const int node0 = (blockIdx.x * (blockDim.x >> 5) + wave) * 16;
  if (node0 >= n) return;

  const int row = node0 + lm;
  const float c0 = b1[lm];
  const float c1 = b1[lm + 16];
  v8f acc0 = {c0, c0, c0, c0, c0, c0, c0, c0};
  v8f acc1 = {c1, c1, c1, c1, c1, c1, c1, c1};

#pragma unroll
  for (int k0 = 0; k0 < HID; k0 += 4) {
    const int ka = k0 + 2 * half;
    v2f a = *(const v2f*)(h + (size_t)row * HID + ka);
    v2f b0 = {W1[ka * HID + lm],      W1[(ka + 1) * HID + lm]};
    v2f b1v = {W1[ka * HID + lm + 16], W1[(ka + 1) * HID + lm + 16]};
    acc0 = wmma_f32(a, b0, acc0);
    acc1 = wmma_f32(a, b1v, acc1);
  }

#pragma unroll
  for (int v = 0; v < 8; v++) {
    const int m = v + 8 * half;
    tbuf[wave][m * HID + lm]      = fmaxf(acc0[v], 0.0f);
    tbuf[wave][m * HID + lm + 16] = fmaxf(acc1[v], 0.0f);
  }
  // same-wave LDS is in-order; wait for ds stores before cross-lane reads
  asm volatile("s_wait_dscnt 0" ::: "memory");

  if (lane < 16) {
    float o0 = b2[0], o1 = b2[1], o2 = b2[2];
#pragma unroll
    for (int k = 0; k < HID; k++) {
      const float t = tbuf[wave][lm * HID + k];
      o0 = fmaf(t, W2[k * 3 + 0], o0);
      o1 = fmaf(t, W2[k * 3 + 1], o1);
      o2 = fmaf(t, W2[k * 3 + 2], o2);
    }
    const size_t o = (size_t)(node0 + lm) * 3;
    out[o + 0] = o0;
    out[o + 1] = o1;
    out[o + 2] = o2;
  }
}

// ---------------------------------------------------------------------------
extern "C" void kernel_launch(void* const* d_in, const int* in_sizes, int n_in,
                              void* d_out, int out_size, void* d_ws,
                              size_t ws_size, hipStream_t stream) {
  const float* x   = (const float*)d_in[0];
  const int*  edge = (const int*)d_in[1];  // [2, E] int32
  const float* Wl1 = (const float*)d_in[2];
  const float* bl1 = (const float*)d_in[3];
  const float* Wr1 = (const float*)d_in[4];
  const float* Wl2 = (const float*)d_in[5];
  const float* bl2 = (const float*)d_in[6];
  const float* Wr2 = (const float*)d_in[7];
  const float* Wl3 = (const float*)d_in[8];
  const float* bl3 = (const float*)d_in[9];
  const float* Wr3 = (const float*)d_in[10];
  const float* Wh1 = (const float*)d_in[11];
  const float* bh1 = (const float*)d_in[12];
  const float* Wh2 = (const float*)d_in[13];
  const float* bh2 = (const float*)d_in[14];
  float* out = (float*)d_out;

  const int N = in_sizes[0] / 8;  // NODE_DIM = 8
  const int E = in_sizes[1] / 2;
  const int* src = edge;
  const int* dst = edge + E;

  float* ws   = (float*)d_ws;
  float* msg  = ws;                      // N*32 floats
  float* ideg = msg + (size_t)N * 32;    // N floats
  float* hA   = ideg + N;                // N*32 floats
  float* hB   = hA + (size_t)N * 32;     // N*32 floats

  const int T = 256;
  const int gemmBlocks = (((N + 15) / 16) + 7) / 8;  // 8 waves/block, 16 nodes/wave

  // degrees -> invdeg
  hipMemsetAsync(ideg, 0, (size_t)N * sizeof(float), stream);
  degree_kernel<<<(E + T - 1) / T, T, 0, stream>>>(dst, ideg, E);
  invdeg_kernel<<<(N + T - 1) / T, T, 0, stream>>>(ideg, N);

  // layer 1: din = 8
  hipMemsetAsync(msg, 0, (size_t)N * 8 * sizeof(float), stream);
  scatter_kernel<8><<<(int)(((size_t)E * 8 + T - 1) / T), T, 0, stream>>>(
      x, src, dst, msg, E);
  sage_gemm_kernel<8><<<gemmBlocks, T, 0, stream>>>(x, msg, ideg, Wl1, bl1,
                                                    Wr1, hA, N);

  // layer 2: din = 32
  hipMemsetAsync(msg, 0, (size_t)N * 32 * sizeof(float), stream);
  scatter_kernel<32><<<(int)(((size_t)E * 32 + T - 1) / T), T, 0, stream>>>(
      hA, src, dst, msg, E);
  sage_gemm_kernel<32><<<gemmBlocks, T, 0, stream>>>(hA, msg, ideg, Wl2, bl2,
                                                     Wr2, hB, N);

  // layer 3: din = 32
  hipMemsetAsync(msg, 0, (size_t)N * 32 * sizeof(float), stream);
  scatter_kernel<32><<<(int)(((size_t)E * 32 + T - 1) / T), T, 0, stream>>>(
      hB, src, dst, msg, E);
  sage_gemm_kernel<32><<<gemmBlocks, T, 0, stream>>>(hB, msg, ideg, Wl3, bl3,
                                                     Wr3, hA, N);

  // MLP head
  head_kernel<<<gemmBlocks, T, 0, stream>>>(hA, Wh1, bh1, Wh2, bh2, out, N);
}